// RWKVRegionCell_43696997270139
// MI455X (gfx1250) — compile-verified
//
#include <hip/hip_runtime.h>

#define DD 4096
#define KSEL 512

typedef __attribute__((ext_vector_type(2))) float v2f;
typedef __attribute__((ext_vector_type(8))) float v8f;

// ---------------------------------------------------------------------------
// K1: x = (x_in + pred) * rsqrt(mean(sq)+eps) * g      (one workgroup)
// ---------------------------------------------------------------------------
__global__ __launch_bounds__(256) void prep_x_kernel(const float* __restrict__ x_in,
                                                     const float* __restrict__ pred,
                                                     const float* __restrict__ g,
                                                     float* __restrict__ xout) {
  __shared__ float buf[DD];
  __shared__ float red[256];
  __shared__ float inv;
  float s = 0.f;
  for (int i = threadIdx.x; i < DD; i += 256) {
    float t = x_in[i] + pred[i];
    buf[i] = t;
    s += t * t;
  }
  red[threadIdx.x] = s;
  __syncthreads();
  if (threadIdx.x == 0) {
    float tot = 0.f;
    for (int i = 0; i < 256; ++i) tot += red[i];   // fixed order -> deterministic
    inv = rsqrtf(tot * (1.0f / DD) + 1e-8f);
  }
  __syncthreads();
  float iv = inv;
  for (int i = threadIdx.x; i < DD; i += 256) xout[i] = buf[i] * iv * g[i];
}

// ---------------------------------------------------------------------------
// Shared WMMA matvec body: out[m0..m0+15] = W[16 rows] @ vec (+ residual)
//   block = 256 threads = 8 waves; each wave owns a 512-wide K chunk; LDS reduce.
//   A (16x4 f32): lane L<16 holds W[row,k..k+1]; lane L+16 holds W[row,k+2..k+3]
//   B (4x16 f32): x chunk broadcast to all 16 columns -> D[m,n] = partial y[m], all n
// ---------------------------------------------------------------------------
__device__ __forceinline__ void matvec16_body(const float* __restrict__ W,
                                              const float* __restrict__ vec,
                                              float* __restrict__ out,
                                              const float* __restrict__ residual,
                                              int m0,
                                              float* xs /* [DD] LDS */,
                                              float (*part)[16] /* [8][16] LDS */) {
  const int tid  = threadIdx.x;
  const int lane = tid & 31;
  const int wv   = tid >> 5;

  // stage the vector in LDS (16 KB)
  for (int i = tid; i < DD / 4; i += 256)
    ((float4*)xs)[i] = ((const float4*)vec)[i];
  __syncthreads();

#if defined(__AMDGCN__)
  const int off = (lane < 16) ? 0 : 2;                      // cndmask, no divergence
  const float* wrow = W + (size_t)(m0 + (lane & 15)) * DD + off;
  v8f c0 = {0.f, 0.f, 0.f, 0.f, 0.f, 0.f, 0.f, 0.f};
  v8f c1 = {0.f, 0.f, 0.f, 0.f, 0.f, 0.f, 0.f, 0.f};
  const int kbeg = wv << 9;                                 // wave's 512-wide K chunk
#pragma unroll 4
  for (int k = kbeg; k < kbeg + 512; k += 8) {
    v2f a0 = *(const v2f*)(wrow + k);                       // global_load_b64 of W
    v2f b0 = *(const v2f*)(xs + k + off);                   // ds_load (x pair)
    v2f a1 = *(const v2f*)(wrow + k + 4);
    v2f b1 = *(const v2f*)(xs + k + 4 + off);
#if __has_builtin(__builtin_amdgcn_wmma_f32_16x16x4_f32)
    c0 = __builtin_amdgcn_wmma_f32_16x16x4_f32(false, a0, false, b0, (short)0, c0, false, false);
    c1 = __builtin_amdgcn_wmma_f32_16x16x4_f32(false, a1, false, b1, (short)0, c1, false, false);
#else
    asm volatile("v_wmma_f32_16x16x4_f32 %0, %1, %2, %0" : "+v"(c0) : "v"(a0), "v"(b0));
    asm volatile("v_wmma_f32_16x16x4_f32 %0, %1, %2, %0" : "+v"(c1) : "v"(a1), "v"(b1));
#endif
  }
  // D layout: VGPR r holds row r (lanes 0-15) and row r+8 (lanes 16-31); all cols equal.
  if (lane == 0) {
#pragma unroll
    for (int m = 0; m < 8; ++m) part[wv][m] = c0[m] + c1[m];
  } else if (lane == 16) {
#pragma unroll
    for (int m = 0; m < 8; ++m) part[wv][8 + m] = c0[m] + c1[m];
  }
#endif  // __AMDGCN__
  __syncthreads();
  if (tid < 16) {
    float s = 0.f;
    for (int w = 0; w < 8; ++w) s += part[w][tid];          // fixed order
    const int row = m0 + tid;
    if (residual) s += residual[row];
    out[row] = s;
  }
}

// Fused r/k/v matvec: 768 blocks, blockIdx>>8 selects the matrix.
__global__ __launch_bounds__(256) void matvec_rkv_kernel(const float* __restrict__ Wr,
                                                         const float* __restrict__ Wk,
                                                         const float* __restrict__ Wv,
                                                         const float* __restrict__ x,
                                                         float* __restrict__ rr,
                                                         float* __restrict__ kr,
                                                         float* __restrict__ vr) {
  __shared__ float xs[DD];
  __shared__ float part[8][16];
  const int mat  = blockIdx.x >> 8;     // 0,1,2
  const int tile = blockIdx.x & 255;
  const float* W = (mat == 0) ? Wr : ((mat == 1) ? Wk : Wv);
  float* out     = (mat == 0) ? rr : ((mat == 1) ? kr : vr);
  matvec16_body(W, x, out, nullptr, tile << 4, xs, part);
}

// Single matvec with residual (W_o pass): h = residual + W @ vec
__global__ __launch_bounds__(256) void matvec16_kernel(const float* __restrict__ W,
                                                       const float* __restrict__ vec,
                                                       float* __restrict__ out,
                                                       const float* __restrict__ residual) {
  __shared__ float xs[DD];
  __shared__ float part[8][16];
  matvec16_body(W, vec, out, residual, blockIdx.x << 4, xs, part);
}

// ---------------------------------------------------------------------------
// K3: r=sigmoid(r_raw); RoPE on v[0:32]; EMA; y = r*num/(den+eps)
// ---------------------------------------------------------------------------
__global__ __launch_bounds__(256) void ema_y_kernel(const float* __restrict__ r_raw,
                                                    const float* __restrict__ k_raw,
                                                    const float* __restrict__ v_raw,
                                                    const float* __restrict__ snum,
                                                    const float* __restrict__ sden,
                                                    const float* __restrict__ rawdec,
                                                    const float* __restrict__ wtime,
                                                    const float* __restrict__ spos,
                                                    float* __restrict__ y) {
  const int idx = blockIdx.x * 256 + threadIdx.x;
  if (idx >= DD) return;
  // lam = max(exp(max(-softplus(raw), -30)), eps)
  float z   = rawdec[idx];
  float sp  = fmaxf(z, 0.f) + log1pf(expf(-fabsf(z)));
  float lam = fmaxf(expf(fmaxf(-sp, -30.f)), 1e-8f);

  float v = v_raw[idx];
  if (idx < 32) {                       // RoPE pairs on first 2*M_TIME channels
    int i = idx >> 1;
    float th = spos[0] * wtime[i];
    float c = cosf(th), s = sinf(th);
    float v0 = v_raw[2 * i], v1 = v_raw[2 * i + 1];
    v = (idx & 1) ? (v0 * s + v1 * c) : (v0 * c - v1 * s);
  }
  float w   = expf(fminf(k_raw[idx], 30.f));
  float num = snum[idx] * lam + w * v;
  float den = sden[idx] * lam + w;
  float r   = 1.f / (1.f + expf(-r_raw[idx]));
  y[idx] = r * (num / (den + 1e-8f));
}

// ---------------------------------------------------------------------------
// K5: exact KWTA threshold (512-th largest) via MSB-first radix select on
//     monotonic float keys; then h = (h >= thr) ? h : 0.  One workgroup.
// ---------------------------------------------------------------------------
__global__ __launch_bounds__(256) void kwta_kernel(float* __restrict__ h) {
  __shared__ unsigned keys[DD];
  __shared__ unsigned cnt;
  const int tid = threadIdx.x;
  for (int i = tid; i < DD; i += 256) {
    unsigned u = __float_as_uint(h[i]);
    keys[i] = (u & 0x80000000u) ? ~u : (u | 0x80000000u);   // order-preserving key
  }
  __syncthreads();

  unsigned prefix = 0;
  unsigned kk = KSEL;
  for (int b = 31; b >= 0; --b) {
    const unsigned bit = 1u << b;
    const unsigned hi  = ~((bit << 1) - 1u);                // bits strictly above b
    if (tid == 0) cnt = 0;
    __syncthreads();
    unsigned local = 0;
    for (int i = tid; i < DD; i += 256) {
      unsigned key = keys[i];
      if (((key ^ prefix) & hi) == 0u && (key & bit)) ++local;
    }
    if (local) atomicAdd(&cnt, local);                      // integer -> deterministic
    __syncthreads();
    unsigned c = cnt;
    __syncthreads();                                        // protect cnt before reset
    if (c >= kk) prefix |= bit; else kk -= c;               // uniform update
  }
  // prefix == key of the 512-th largest value; keep key >= prefix
  for (int i = tid; i < DD; i += 256)
    if (keys[i] < prefix) h[i] = 0.0f;
}

// ---------------------------------------------------------------------------
extern "C" void kernel_launch(void* const* d_in, const int* in_sizes, int n_in,
                              void* d_out, int out_size, void* d_ws, size_t ws_size,
                              hipStream_t stream) {
  (void)in_sizes; (void)n_in; (void)out_size; (void)ws_size;
  const float* x_in      = (const float*)d_in[0];
  const float* pred      = (const float*)d_in[1];
  const float* state_num = (const float*)d_in[2];
  const float* state_den = (const float*)d_in[3];
  const float* g_norm    = (const float*)d_in[4];
  const float* W_r       = (const float*)d_in[5];
  const float* W_k       = (const float*)d_in[6];
  const float* W_v       = (const float*)d_in[7];
  const float* W_o       = (const float*)d_in[8];
  const float* raw_decay = (const float*)d_in[9];
  const float* W_time    = (const float*)d_in[10];
  const float* step_pos  = (const float*)d_in[11];

  float* ws = (float*)d_ws;
  float* x  = ws;               // [0,      4096)
  float* rr = ws + DD;          // [4096,   8192)
  float* kr = ws + 2 * DD;      // [8192,  12288)
  float* vr = ws + 3 * DD;      // [12288, 16384)
  float* y  = ws + 4 * DD;      // [16384, 20480)
  float* h  = (float*)d_out;    // 4096 outputs

  prep_x_kernel<<<1, 256, 0, stream>>>(x_in, pred, g_norm, x);
  // fused r/k/v weight streaming: 768 blocks -> ~3 blocks per WGP in flight
  matvec_rkv_kernel<<<3 * (DD / 16), 256, 0, stream>>>(W_r, W_k, W_v, x, rr, kr, vr);
  ema_y_kernel<<<DD / 256, 256, 0, stream>>>(rr, kr, vr, state_num, state_den,
                                             raw_decay, W_time, step_pos, y);
  matvec16_kernel<<<DD / 16, 256, 0, stream>>>(W_o, y, h, x);  // h = x + W_o@y
  kwta_kernel<<<1, 256, 0, stream>>>(h);
}